// BNNLinear_39934605918275
// MI455X (gfx1250) — compile-verified
//
#include <hip/hip_runtime.h>
#include <math.h>

typedef float v2f __attribute__((ext_vector_type(2)));
typedef float v8f __attribute__((ext_vector_type(8)));

// 4-float vector with only 4-byte alignment guarantee (E/s rows have stride
// 513 floats = 2052 B, so row bases are only DWORD-aligned).
typedef float f32x4 __attribute__((ext_vector_type(4)));
typedef f32x4 f32x4u __attribute__((aligned(4)));

static constexpr int kM = 256;   // samples
static constexpr int kC = 512;   // out_dim
static constexpr int kIn = 512;  // in_dim
static constexpr int kR = 513;   // in_dim + bias row

// ---------------------------------------------------------------------------
// Kernel 0: s = sqrt(var), 512*513 elements into workspace (L2-resident).
// ---------------------------------------------------------------------------
__global__ void __launch_bounds__(256) sqrt_kernel(const float* __restrict__ var,
                                                   float* __restrict__ s, int n) {
    int i = blockIdx.x * 256 + threadIdx.x;
    if (i < n) s[i] = sqrtf(var[i]);
}

// ---------------------------------------------------------------------------
// Kernel 1: h1 = x_ext @ mu^T via V_WMMA_F32_16X16X4_F32 (fp32-exact).
// One wave32 per 16x16 output tile. A 16x4 f32 fragment layout (ISA 7.12.2):
//   lane L: M = L&15 ; VGPR0/1 hold K = 2*(L>>4), 2*(L>>4)+1
// B 4x16 is the symmetric row-striped layout: lane L: N = L&15, same K pair.
// C/D: VGPR v -> (M = v + 8*(L>>4), N = L&15).
// ---------------------------------------------------------------------------
__global__ void __launch_bounds__(32) h1_wmma_kernel(const float* __restrict__ x,
                                                     const float* __restrict__ mu,
                                                     float* __restrict__ out) {
    const int lane = threadIdx.x;
    const int lo   = lane & 15;
    const int hi   = lane >> 4;          // 0 or 1
    const int c0   = blockIdx.x * 16;
    const int m0   = blockIdx.y * 16;

    const float* xr = x  + (size_t)(m0 + lo) * kIn;  // A-matrix row (M = lo)
    const float* mr = mu + (size_t)(c0 + lo) * kR;   // B-matrix col (N = lo) = mu row

    v8f acc = {};
    const int ks = 2 * hi;
#pragma unroll 4
    for (int k = 0; k < kIn; k += 4) {
        v2f a, b;
        a.x = xr[k + ks];
        a.y = xr[k + ks + 1];
        b.x = mr[k + ks];
        b.y = mr[k + ks + 1];
        // 8 args: (neg_a, A, neg_b, B, c_mod, C, reuse_a, reuse_b)
        acc = __builtin_amdgcn_wmma_f32_16x16x4_f32(
            false, a, false, b, (short)0, acc, false, false);
    }

    const float bias = mr[kIn];  // x_ext[m,512] == 1 -> + mu[c,512]
#pragma unroll
    for (int v = 0; v < 8; ++v) {
        const int m = m0 + 8 * hi + v;
        out[(size_t)m * kC + (c0 + lo)] = acc[v] + bias;
    }
}

// ---------------------------------------------------------------------------
// Kernel 2: out[m,c] += sum_r x_ext[m,r] * s[c,r] * E[m,c,r]
// One wave32 per output element; lanes split r in float4 chunks so the wave
// reads 512 contiguous bytes of E per step (fully coalesced HBM stream).
// E (269 MB, single pass, > 192 MB L2) is loaded with TH=NT so it does not
// evict the hot, heavily-reused x/s/mu working set from L2.
// ---------------------------------------------------------------------------
__global__ void __launch_bounds__(256) h2_kernel(const float* __restrict__ x,
                                                 const float* __restrict__ s,
                                                 const float* __restrict__ E,
                                                 float* __restrict__ out,
                                                 int precomputed) {
    const int lane = threadIdx.x & 31;
    const int wid  = blockIdx.x * (blockDim.x >> 5) + (threadIdx.x >> 5);
    const int m = wid >> 9;      // / 512
    const int c = wid & 511;

    const float* xr = x + (size_t)m * kIn;
    const float* sr = s + (size_t)c * kR;
    const float* er = E + (size_t)wid * kR;   // (m*512 + c)*513

    float acc = 0.0f;
#pragma unroll
    for (int i = 0; i < 4; ++i) {
        const int r = (lane + 32 * i) * 4;    // 0..508, covers r = 0..511
        // Non-temporal, align-4 b128 stream of E
        const f32x4 e4 =
            __builtin_nontemporal_load(reinterpret_cast<const f32x4u*>(er + r));
        const f32x4 s4 = *reinterpret_cast<const f32x4u*>(sr + r);
        const f32x4 x4 = *reinterpret_cast<const f32x4u*>(xr + r);
#pragma unroll
        for (int k = 0; k < 4; ++k) {
            const float sv = precomputed ? s4[k] : sqrtf(s4[k]);
            acc = fmaf(x4[k] * sv, e4[k], acc);
        }
    }
    if (lane == 0) {  // bias row r = 512: x_ext = 1
        const float sv = precomputed ? sr[kIn] : sqrtf(sr[kIn]);
        acc = fmaf(sv, __builtin_nontemporal_load(er + kIn), acc);
    }

    // wave32 shuffle reduction
#pragma unroll
    for (int off = 16; off > 0; off >>= 1)
        acc += __shfl_down(acc, off, 32);

    if (lane == 0) out[wid] += acc;   // h1 already written by kernel 1
}

// ---------------------------------------------------------------------------
extern "C" void kernel_launch(void* const* d_in, const int* in_sizes, int n_in,
                              void* d_out, int out_size, void* d_ws, size_t ws_size,
                              hipStream_t stream) {
    const float* x   = (const float*)d_in[0];   // [256, 512]
    const float* mu  = (const float*)d_in[1];   // [512, 513]
    const float* var = (const float*)d_in[2];   // [512, 513]
    const float* E   = (const float*)d_in[3];   // [256, 512, 513]
    float* out = (float*)d_out;                 // [256, 512]
    (void)in_sizes; (void)n_in; (void)out_size;

    const int nvs = kC * kR;  // 262656
    float* s = (float*)d_ws;
    const int pre = (ws_size >= (size_t)nvs * sizeof(float)) ? 1 : 0;

    if (pre) {
        sqrt_kernel<<<(nvs + 255) / 256, 256, 0, stream>>>(var, s, nvs);
    }

    // h1: 32 c-tiles x 16 m-tiles, one wave each
    h1_wmma_kernel<<<dim3(kC / 16, kM / 16), 32, 0, stream>>>(x, mu, out);

    // h2: one wave per output element, 8 waves per block
    const int nwaves = kM * kC;  // 131072
    h2_kernel<<<nwaves / 8, 256, 0, stream>>>(x, pre ? s : var, E, out, pre);
}